// UnifiedSAS_9612136808526
// MI455X (gfx1250) — compile-verified
//
#include <hip/hip_runtime.h>
#include <hip/hip_bf16.h>
#include <math.h>

// ---------------------------------------------------------------------------
// Problem constants (x: [32,3,256,256] f32)
// ---------------------------------------------------------------------------
#define SAS_B 32
#define SAS_C 3
#define SAS_H 256
#define SAS_W 256
#define SAS_NCH (SAS_B * SAS_C)                     // 96 independent images
#define SAS_IMG (SAS_H * SAS_W)                     // 65536
#define SAS_PERB (SAS_C * SAS_H * SAS_W)            // 196608 (median domain)
#define SAS_TOTAL (SAS_B * SAS_C * SAS_H * SAS_W)   // 6291456

#define TILE 64
#define HALO 3                 // 3 TV iterations -> halo of 3
#define LW (TILE + 2 * HALO)   // 70
#define LSTR (LW + 1)          // 71 (odd stride -> fewer LDS bank conflicts)
#define NPOS (LW * LW)         // 4900
#define NCTR (TILE * TILE)     // 4096
#define NT 256                 // 8 wave32 per workgroup

#define SAS_STEP 0.2f

// ---------------------------------------------------------------------------
// CDNA5 async global<->LDS copies (ASYNCcnt path, CDNA5 ISA
// §4).  Probe-verified: builtin pointer params are int* in AS1 / AS3, order
// (global, lds, imm offset, imm cpol).  Guarded with plain-load fallbacks.
// ---------------------------------------------------------------------------
#if defined(__has_builtin)
#  if __has_builtin(__builtin_amdgcn_global_load_async_to_lds_b32)
#    define SAS_HAVE_ASYNC_LDS 1
#  endif
#  if __has_builtin(__builtin_amdgcn_global_store_async_from_lds_b32)
#    define SAS_HAVE_ASYNC_ST 1
#  endif
#endif

#if defined(SAS_HAVE_ASYNC_LDS) || defined(SAS_HAVE_ASYNC_ST)
typedef __attribute__((address_space(1))) int* sas_gptr_t;
typedef __attribute__((address_space(3))) int* sas_lptr_t;
#endif

__device__ __forceinline__ void sas_async_load_f32(const float* g, float* l) {
#ifdef SAS_HAVE_ASYNC_LDS
  __builtin_amdgcn_global_load_async_to_lds_b32(
      (sas_gptr_t)(void*)const_cast<float*>(g),
      (sas_lptr_t)(void*)l,
      /*offset=*/0, /*cpol=*/0);
#else
  *l = *g;
#endif
}

__device__ __forceinline__ void sas_async_store_f32(float* g, const float* l) {
#ifdef SAS_HAVE_ASYNC_ST
  __builtin_amdgcn_global_store_async_from_lds_b32(
      (sas_gptr_t)(void*)g,
      (sas_lptr_t)(void*)const_cast<float*>(l),
      /*offset=*/0, /*cpol=*/0);
#else
  *g = *l;
#endif
}

__device__ __forceinline__ void sas_async_join() {
#if defined(SAS_HAVE_ASYNC_LDS) || defined(SAS_HAVE_ASYNC_ST)
#  if __has_builtin(__builtin_amdgcn_s_wait_asynccnt)
  __builtin_amdgcn_s_wait_asynccnt(0);
#  else
  asm volatile("s_wait_asynccnt 0" ::: "memory");
#  endif
#endif
}

// ---------------------------------------------------------------------------
// ISA probe kernel (NEVER LAUNCHED).  Placed first in the file so the disasm
// snippet (which shows the first function) displays the async-tensor
// mnemonics plainly.  Harmless: kernel_launch never enqueues it.
// ---------------------------------------------------------------------------
__global__ void sas_isa_probe_kernel(const float* __restrict__ g,
                                     float* __restrict__ o) {
  __shared__ float l[NT];
  const int t = threadIdx.x;
  sas_async_load_f32(g + t, &l[t]);   // -> global_load_async_to_lds_b32
  sas_async_join();                   // -> s_wait_asynccnt 0
  __syncthreads();
  sas_async_store_f32(o + t, &l[t]);  // -> global_store_async_from_lds_b32
  sas_async_join();
}

// ---------------------------------------------------------------------------
// Median-of-5 via optimal 9-comparator sorting network (Knuth):
// (0,1)(3,4)(2,4)(2,3)(0,3)(0,2)(1,4)(1,3)(1,2); a2 is the median.
// jnp.median over K=5 (odd) == the middle order statistic.
// ---------------------------------------------------------------------------
__device__ __forceinline__ void sas_cswap(float& a, float& b) {
  float lo = fminf(a, b), hi = fmaxf(a, b);
  a = lo; b = hi;
}
__device__ __forceinline__ float sas_med5(float a0, float a1, float a2, float a3, float a4) {
  sas_cswap(a0, a1); sas_cswap(a3, a4); sas_cswap(a2, a4);
  sas_cswap(a2, a3); sas_cswap(a0, a3); sas_cswap(a0, a2);
  sas_cswap(a1, a4); sas_cswap(a1, a3); sas_cswap(a1, a2);
  return a2;
}

// ---------------------------------------------------------------------------
// K2: fused Huber-TV (4 views x 3 iterations) per 64x64 tile with halo 3.
// Boundary conditions use TRUE global coords (dx padded 0 at j==W-1; the
// divergence at j==0 omits gx(j-1)), so valid-region shrinks by 1 ring per
// iteration; center 64x64 stays exact.  Also emits per-tile sum of per-pixel
// MAD (needed later for the global mad.mean() floor) deterministically.
// LDS: 3*4970 + 5*4096 + 256 floats = 142,584 B (accepted by gfx1250 backend).
// ---------------------------------------------------------------------------
__global__ __launch_bounds__(NT) void sas_tv_kernel(const float* __restrict__ x,
                                                    const float* __restrict__ scale,
                                                    float* __restrict__ arts,
                                                    float* __restrict__ partials,
                                                    float4 lams, float4 dels) {
  __shared__ float xn[LW * LSTR];       // normalized input tile (x / scale)
  __shared__ float buf[2][LW * LSTR];   // ping-pong diffusion buffers
  __shared__ float orig[NCTR];          // exact raw x for the center tile
  __shared__ float artc[4][NCTR];       // center-tile denoised*scale per view
  __shared__ float red[NT];

  const int tx = blockIdx.x, ty = blockIdx.y, bc = blockIdx.z;
  const int tid = threadIdx.x;
  const float s = scale[bc / SAS_C];
  const int gx0 = tx * TILE - HALO, gy0 = ty * TILE - HALO;
  const float* img = x + (size_t)bc * SAS_IMG;

  // --- async tile load (coords clamped; out-of-image halo values are never
  //     consumed by valid pixels thanks to the BCs) -------------------------
  for (int p = tid; p < NPOS; p += NT) {
    int li = p / LW, lj = p - li * LW;
    int gi = min(max(gy0 + li, 0), SAS_H - 1);
    int gj = min(max(gx0 + lj, 0), SAS_W - 1);
    sas_async_load_f32(img + gi * SAS_W + gj, &xn[li * LSTR + lj]);
  }
  sas_async_join();
  __syncthreads();

  // --- save exact originals for center, normalize in place ----------------
  for (int p = tid; p < NPOS; p += NT) {
    int li = p / LW, lj = p - li * LW;
    int idx = li * LSTR + lj;
    float raw = xn[idx];
    int ci = li - HALO, cj = lj - HALO;
    if (ci >= 0 && ci < TILE && cj >= 0 && cj < TILE) orig[ci * TILE + cj] = raw;
    xn[idx] = raw / s;
  }
  __syncthreads();

  const float lamv[4] = {lams.x, lams.y, lams.z, lams.w};
  const float delv[4] = {dels.x, dels.y, dels.z, dels.w};

  for (int view = 0; view < 4; ++view) {
    const float lam = lamv[view];
    const float delta = delv[view];
    // xd starts from xn every view
    for (int p = tid; p < NPOS; p += NT) {
      int li = p / LW;
      buf[0][p + li] = xn[p + li];
    }
    __syncthreads();
    int cb = 0;
    for (int it = 0; it < 3; ++it) {
      for (int p = tid; p < NPOS; p += NT) {
        int li = p / LW, lj = p - li * LW;
        int idx = li * LSTR + lj;
        int gi = gy0 + li, gj = gx0 + lj;  // true (unclamped) coords for BCs
        float c   = buf[cb][idx];
        float xr  = buf[cb][li * LSTR + min(lj + 1, LW - 1)];
        float xl  = buf[cb][li * LSTR + max(lj - 1, 0)];
        float xdn = buf[cb][min(li + 1, LW - 1) * LSTR + lj];
        float xup = buf[cb][max(li - 1, 0) * LSTR + lj];
        float dxp = (gj < SAS_W - 1) ? (xr - c) : 0.f;
        float dyp = (gi < SAS_H - 1) ? (xdn - c) : 0.f;
        float gxp = dxp / (delta + fabsf(dxp));
        float gyp = dyp / (delta + fabsf(dyp));
        float gxm = 0.f, gym = 0.f;
        if (gj > 0) { float d = c - xl;  gxm = d / (delta + fabsf(d)); }
        if (gi > 0) { float d = c - xup; gym = d / (delta + fabsf(d)); }
        float divg = (gxp - gxm) + (gyp - gym);
        buf[cb ^ 1][idx] = c - SAS_STEP * (c - xn[idx] - lam * divg);
      }
      __syncthreads();
      cb ^= 1;
    }
    // center results -> LDS stash for the MAD pass (ds_store),
    // then barrier (waits DScnt) so the async DMA engine sees the data,
    // then DMA the staged tile out to global (overlaps next view's stencil).
    for (int q = tid; q < NCTR; q += NT) {
      int ci = q / TILE, cj = q - ci * TILE;
      artc[view][q] = buf[cb][(ci + HALO) * LSTR + (cj + HALO)] * s;
    }
    __syncthreads();
    {
      float* dstv = arts + (size_t)view * SAS_TOTAL + (size_t)bc * SAS_IMG;
      for (int q = tid; q < NCTR; q += NT) {
        int ci = q / TILE, cj = q - ci * TILE;
        sas_async_store_f32(dstv + (size_t)(ty * TILE + ci) * SAS_W + (tx * TILE + cj),
                            &artc[view][q]);
      }
    }
  }

  // --- deterministic per-tile sum of per-pixel MAD ------------------------
  float acc = 0.f;
  for (int q = tid; q < NCTR; q += NT) {
    float a0 = orig[q], a1 = artc[0][q], a2 = artc[1][q], a3 = artc[2][q], a4 = artc[3][q];
    float med = sas_med5(a0, a1, a2, a3, a4);
    acc += sas_med5(fabsf(a0 - med), fabsf(a1 - med), fabsf(a2 - med),
                    fabsf(a3 - med), fabsf(a4 - med));
  }
  red[tid] = acc;
  __syncthreads();
  for (int off = NT / 2; off > 0; off >>= 1) {
    if (tid < off) red[tid] += red[tid + off];
    __syncthreads();
  }
  if (tid == 0)
    partials[((size_t)bc * (SAS_H / TILE) + ty) * (SAS_W / TILE) + tx] = red[0];
  sas_async_join();  // drain outstanding DMA stores (s_endpgm also waits idle)
}

// ---------------------------------------------------------------------------
// K1: exact per-batch lower-median of |x| via MSB-first radix select.
// One workgroup per batch; 4 passes of 256-bin LDS histograms.
// |x| as uint bits (sign cleared) is order-preserving for floats >= 0.
// Target rank k = (196608-1)//2 = 98303 (0-based k-th smallest).
// Passes 2..4 re-read the batch from the 192 MB L2 (whole tensor resident).
// ---------------------------------------------------------------------------
__global__ __launch_bounds__(NT) void sas_median_kernel(const float* __restrict__ x,
                                                        float* __restrict__ scale) {
  const int b = blockIdx.x;
  const int tid = threadIdx.x;
  const float* base = x + (size_t)b * SAS_PERB;
  __shared__ unsigned hist[256];
  __shared__ unsigned sh_prefix;
  __shared__ unsigned sh_want;
  if (tid == 0) { sh_prefix = 0u; sh_want = (SAS_PERB - 1) / 2; }
  __syncthreads();
  for (int pass = 0; pass < 4; ++pass) {
    const int shift = 24 - 8 * pass;
    hist[tid] = 0u;  // NT == 256
    __syncthreads();
    const unsigned prefix = sh_prefix;
    for (int i = tid; i < SAS_PERB; i += NT) {
      unsigned u = __float_as_uint(base[i]) & 0x7fffffffu;
      // match already-selected higher bytes (uint64 shift: safe when shift+8==32)
      if ((((unsigned long long)(u ^ prefix)) >> (shift + 8)) == 0ull)
        atomicAdd(&hist[(u >> shift) & 255u], 1u);
    }
    __syncthreads();
    if (tid == 0) {
      unsigned want = sh_want, cum = 0u;
      for (int d = 0; d < 256; ++d) {
        unsigned c = hist[d];
        if (cum + c > want) {
          sh_prefix = prefix | ((unsigned)d << shift);
          sh_want = want - cum;
          break;
        }
        cum += c;
      }
    }
    __syncthreads();
  }
  if (tid == 0) scale[b] = __uint_as_float(sh_prefix) + 1e-8f;
}

// ---------------------------------------------------------------------------
// K2b: fixed-order reduction of tile partials -> mad_mean (deterministic).
// ---------------------------------------------------------------------------
__global__ __launch_bounds__(NT) void sas_reduce_kernel(const float* __restrict__ partials,
                                                        int n,
                                                        float* __restrict__ mad_mean) {
  __shared__ float red[NT];
  float ssum = 0.f;
  for (int i = threadIdx.x; i < n; i += NT) ssum += partials[i];
  red[threadIdx.x] = ssum;
  __syncthreads();
  for (int off = NT / 2; off > 0; off >>= 1) {
    if (threadIdx.x < off) red[threadIdx.x] += red[threadIdx.x + off];
    __syncthreads();
  }
  if (threadIdx.x == 0) *mad_mean = red[0] / (float)SAS_TOTAL;
}

// ---------------------------------------------------------------------------
// K3: robust MAD-weighted fusion, float4-vectorized grid-stride.
// ---------------------------------------------------------------------------
__device__ __forceinline__ float sas_fuse1(float a0, float a1, float a2, float a3,
                                           float a4, float mm01) {
  float med = sas_med5(a0, a1, a2, a3, a4);
  float d0 = fabsf(a0 - med), d1 = fabsf(a1 - med), d2 = fabsf(a2 - med);
  float d3 = fabsf(a3 - med), d4 = fabsf(a4 - med);
  float mad = sas_med5(d0, d1, d2, d3, d4);
  float madf = fmaxf(mad, mm01);
  float inv = 1.0f / (2.0f * madf);  // FUSION_SCALE = 2
  float w0 = __expf(-d0 * inv), w1 = __expf(-d1 * inv), w2 = __expf(-d2 * inv);
  float w3 = __expf(-d3 * inv), w4 = __expf(-d4 * inv);
  float num = w0 * a0 + w1 * a1 + w2 * a2 + w3 * a3 + w4 * a4;
  float den = w0 + w1 + w2 + w3 + w4 + 1e-8f;
  return num / den;
}

__global__ __launch_bounds__(NT) void sas_fuse_kernel(const float* __restrict__ x,
                                                      const float* __restrict__ arts,
                                                      const float* __restrict__ mad_mean,
                                                      float* __restrict__ out) {
  const int total4 = SAS_TOTAL / 4;
  const float mm01 = 0.1f * (*mad_mean);
  const float4* x4 = (const float4*)x;
  const float4* v14 = (const float4*)(arts + 0 * (size_t)SAS_TOTAL);
  const float4* v24 = (const float4*)(arts + 1 * (size_t)SAS_TOTAL);
  const float4* v34 = (const float4*)(arts + 2 * (size_t)SAS_TOTAL);
  const float4* v44 = (const float4*)(arts + 3 * (size_t)SAS_TOTAL);
  float4* o4 = (float4*)out;
  for (int i = blockIdx.x * NT + threadIdx.x; i < total4; i += gridDim.x * NT) {
    float4 a = x4[i], b = v14[i], c = v24[i], d = v34[i], e = v44[i];
    float4 r;
    r.x = sas_fuse1(a.x, b.x, c.x, d.x, e.x, mm01);
    r.y = sas_fuse1(a.y, b.y, c.y, d.y, e.y, mm01);
    r.z = sas_fuse1(a.z, b.z, c.z, d.z, e.z, mm01);
    r.w = sas_fuse1(a.w, b.w, c.w, d.w, e.w, mm01);
    o4[i] = r;
  }
}

// ---------------------------------------------------------------------------
// Launcher.  Workspace layout (bytes):
//   [0,128)        scale[32]
//   [256, +4*T*4)  arts: 4 denoised views, f32 each   (~100.7 MB)
//   then           partials[1536], mad_mean[1]
// ---------------------------------------------------------------------------
extern "C" void kernel_launch(void* const* d_in, const int* in_sizes, int n_in,
                              void* d_out, int out_size, void* d_ws, size_t ws_size,
                              hipStream_t stream) {
  (void)in_sizes; (void)n_in; (void)out_size; (void)ws_size;
  const float* x = (const float*)d_in[0];
  float* out = (float*)d_out;
  char* ws = (char*)d_ws;

  float* scale = (float*)ws;                        // 32 floats
  float* arts = (float*)(ws + 256);                 // 4 * SAS_TOTAL floats
  float* partials = arts + 4 * (size_t)SAS_TOTAL;   // 1536 floats
  float* mad_mean = partials + 2048;                // 1 float

  // Deterministic parameter sets (double math, then cast, matching numpy).
  double lam_d[4], del_d[4];
  for (int i = 0; i < 4; ++i) {
    lam_d[i] = 0.025 + 0.05 * (double)i / 3.0;                 // lmin..lmax
    del_d[i] = 0.01 * sqrt(lam_d[i] / (0.05 + 1e-8));
  }
  float4 lams = make_float4((float)lam_d[0], (float)lam_d[1], (float)lam_d[2], (float)lam_d[3]);
  float4 dels = make_float4((float)del_d[0], (float)del_d[1], (float)del_d[2], (float)del_d[3]);

  sas_median_kernel<<<SAS_B, NT, 0, stream>>>(x, scale);

  dim3 g2(SAS_W / TILE, SAS_H / TILE, SAS_NCH);  // (4, 4, 96)
  sas_tv_kernel<<<g2, NT, 0, stream>>>(x, scale, arts, partials, lams, dels);

  const int npart = (SAS_W / TILE) * (SAS_H / TILE) * SAS_NCH;  // 1536
  sas_reduce_kernel<<<1, NT, 0, stream>>>(partials, npart, mad_mean);

  sas_fuse_kernel<<<1536, NT, 0, stream>>>(x, arts, mad_mean, out);
}